// GraphUnpooling_42099269435630
// MI455X (gfx1250) — compile-verified
//
#include <hip/hip_runtime.h>
#include <hip/hip_bf16.h>

typedef float v2f __attribute__((ext_vector_type(2)));
typedef float v4f __attribute__((ext_vector_type(4)));
typedef float v8f __attribute__((ext_vector_type(8)));

#define HDIM 128
#define KSTEPS 32           // 128 / 4 (K per V_WMMA_F32_16X16X4_F32)
#define NTILES 8            // 128 / 16 output column tiles
#define WAVES_PER_BLOCK 8
#define LN_EPS 1e-5f

// ---------------------------------------------------------------------------
// Kernel 1: h = x @ W^T + b ; LayerNorm(H) ; SiLU  ->  ws  [Mrows x 128] f32
// One wave produces a 16x128 row block via 8 accumulator tiles (v8f each).
// W is staged in LDS pre-permuted into the exact 4x16 B-fragment layout so the
// inner-loop reads are contiguous conflict-free ds_load_b64.
// ---------------------------------------------------------------------------
__global__ __launch_bounds__(256) void fused_linear_ln_silu(
    const float* __restrict__ x, const float* __restrict__ W,
    const float* __restrict__ bias, const float* __restrict__ gamma,
    const float* __restrict__ beta, float* __restrict__ ws, int Mrows)
{
    // [ks][nt][lane] x float2  == 64 KB, a permutation of W
    __shared__ float Wfrag[KSTEPS * NTILES * 32 * 2];

    const int tid = threadIdx.x;

    // Stage W (row-major [o][k], o=output dim, k=input dim) into fragment
    // layout. B-fragment (4x16, K x N): lane l (0-15) holds B[2h][n=l],
    // B[2h+1][n=l] with h = lane>>4, and B[k][n] = W[n][k].
    #pragma unroll
    for (int i = 0; i < (HDIM * HDIM) / 256; ++i) {
        int flat = i * 256 + tid;       // coalesced global read of W
        int o    = flat >> 7;
        int k    = flat & 127;
        int ks   = k >> 2;
        int hf   = (k >> 1) & 1;
        int j    = k & 1;
        int nt   = o >> 4;
        int l15  = o & 15;
        Wfrag[(((ks * NTILES + nt) * 32) + hf * 16 + l15) * 2 + j] = W[flat];
    }
    __syncthreads();

    const int lane = tid & 31;
    const int wave = tid >> 5;
    const int l15  = lane & 15;
    const int half = lane >> 4;
    const int m0   = (blockIdx.x * WAVES_PER_BLOCK + wave) * 16;
    if (m0 >= Mrows) return;   // wave-uniform: EXEC stays all-ones for WMMA

    // Per-lane column params: col = nt*16 + l15 (coalesced 64B loads)
    float bb[NTILES], gg[NTILES], be[NTILES];
    #pragma unroll
    for (int nt = 0; nt < NTILES; ++nt) {
        int col = nt * 16 + l15;
        bb[nt] = bias[col];
        gg[nt] = gamma[col];
        be[nt] = beta[col];
    }

    v8f acc[NTILES];
    #pragma unroll
    for (int nt = 0; nt < NTILES; ++nt)
        #pragma unroll
        for (int r = 0; r < 8; ++r) acc[nt][r] = 0.0f;

    // A-fragment (16x4, M x K): lane l15 holds x[m0+l15][4ks+2h], [4ks+2h+1]
    const float* xr = x + (size_t)(m0 + l15) * HDIM + half * 2;
    const v2f*   wf = (const v2f*)Wfrag;

    #pragma unroll 4
    for (int ks = 0; ks < KSTEPS; ++ks) {
        v2f a = *(const v2f*)(xr + ks * 4);
        #pragma unroll
        for (int nt = 0; nt < NTILES; ++nt) {
            v2f bfr = wf[(ks * NTILES + nt) * 32 + lane];
            acc[nt] = __builtin_amdgcn_wmma_f32_16x16x4_f32(
                false, a, false, bfr, (short)0, acc[nt], false, false);
        }
    }

    // Epilogue: +bias, LayerNorm over the 128 columns each wave fully owns.
    // C/D layout: vgpr r, lanes 0-15 -> row r, lanes 16-31 -> row 8+r.
    float sum[8], ssq[8];
    #pragma unroll
    for (int r = 0; r < 8; ++r) { sum[r] = 0.0f; ssq[r] = 0.0f; }
    #pragma unroll
    for (int nt = 0; nt < NTILES; ++nt)
        #pragma unroll
        for (int r = 0; r < 8; ++r) {
            float v = acc[nt][r] + bb[nt];
            acc[nt][r] = v;
            sum[r] += v;
            ssq[r] += v * v;
        }
    // xor-reduce across the 16 lanes of each half-wave (masks 1,2,4,8)
    #pragma unroll
    for (int m = 1; m <= 8; m <<= 1) {
        #pragma unroll
        for (int r = 0; r < 8; ++r) {
            sum[r] += __shfl_xor(sum[r], m, 32);
            ssq[r] += __shfl_xor(ssq[r], m, 32);
        }
    }

    const float inv = 1.0f / (float)HDIM;
    #pragma unroll
    for (int r = 0; r < 8; ++r) {
        float mean = sum[r] * inv;
        float var  = ssq[r] * inv - mean * mean;
        float rstd = rsqrtf(var + LN_EPS);
        int row = m0 + r + half * 8;
        float* wr = ws + (size_t)row * HDIM + l15;
        #pragma unroll
        for (int nt = 0; nt < NTILES; ++nt) {
            float hn = (acc[nt][r] - mean) * rstd * gg[nt] + be[nt];
            float y  = hn / (1.0f + __expf(-hn));   // SiLU
            wr[nt * 16] = y;
        }
    }
}

// ---------------------------------------------------------------------------
// Kernel 2: out[b, fi, :, :] = ws[b, f2c[fi], :, :]
// One block per (b, fi); Fm*H = 1024 floats = 256 float4 = 1 per thread.
// Gather reads hit L2 (ws = 32 MB << 192 MB L2); the 262 MB output stream is
// written non-temporally so it doesn't evict ws from L2.
// ---------------------------------------------------------------------------
__global__ __launch_bounds__(256) void gather_fine(
    const float* __restrict__ ws, const int* __restrict__ f2c,
    float* __restrict__ out, int NF, int NC)
{
    int blk = blockIdx.x;           // b * NF + fi
    int b   = blk / NF;
    int fi  = blk - b * NF;
    int c   = f2c[fi];              // uniform per block -> scalar load
    const v4f* src = (const v4f*)ws + (((size_t)(b * NC + c)) << 8);
    v4f*       dst = (v4f*)out + ((size_t)blk << 8);
    v4f v = src[threadIdx.x];
    __builtin_nontemporal_store(v, dst + threadIdx.x);
}

extern "C" void kernel_launch(void* const* d_in, const int* in_sizes, int n_in,
                              void* d_out, int out_size, void* d_ws, size_t ws_size,
                              hipStream_t stream) {
    const float* x     = (const float*)d_in[0];   // [B, NC, Fm, H] f32
    const float* W     = (const float*)d_in[1];   // [H, H] f32
    const float* bias  = (const float*)d_in[2];   // [H]
    const float* gamma = (const float*)d_in[3];   // [H]
    const float* beta  = (const float*)d_in[4];   // [H]
    const int*   f2c   = (const int*)d_in[5];     // [NF]

    const int Mrows   = in_sizes[0] / HDIM;       // B*NC*Fm = 64000
    const int NF      = in_sizes[5];              // 16000
    const int outRows = out_size / (8 * HDIM);    // B*NF (Fm = 8 per reference)
    const int Bb      = outRows / NF;             // 4
    const int NC      = (Mrows / Bb) / 8;         // 2000 (Fm = 8)

    float* ws = (float*)d_ws;                     // Mrows*128*4 = 32.8 MB

    const int tilesM  = Mrows / 16;
    const int blocksA = (tilesM + WAVES_PER_BLOCK - 1) / WAVES_PER_BLOCK;  // 500
    fused_linear_ln_silu<<<blocksA, 256, 0, stream>>>(x, W, bias, gamma, beta,
                                                      ws, Mrows);

    gather_fine<<<Bb * NF, 256, 0, stream>>>(ws, f2c, (float*)d_out, NF, NC);
}